// TransposeToMulL_63634235457615
// MI455X (gfx1250) — compile-verified
//
#include <hip/hip_runtime.h>
#include <stdint.h>

// out[row*1024 + i] = features[row*1024 + perm[i]]
// Pure bandwidth problem: ~1.07 GB moved, ~46us floor at 23.3 TB/s.
// Strategy: async-DMA rows into LDS (coalesced b128 global loads, ASYNCcnt,
// non-temporal), permute inside LDS (ds_load_b32 across 64 banks), coalesced
// non-temporal b128 global stores. Double-buffered tiles with partial
// s_wait_asynccnt for DMA/compute overlap.

#define THREADS        256
#define ROW_ELEMS      1024
#define ROW_BYTES      (ROW_ELEMS * 4)
#define ROWS_PER_TILE  4
#define TILE_BYTES     (ROWS_PER_TILE * ROW_BYTES)
#define TILES          8
#define ROWS_PER_BLOCK (ROWS_PER_TILE * TILES)

typedef __attribute__((ext_vector_type(4))) float v4f;

// One async 16B copy, row offset folded into the instruction immediate.
// INST_OFFSET is applied to BOTH the LDS and global addresses; our LDS row
// stride equals the global row stride (4096B), so a single immediate works.
template <int OFF>
__device__ __forceinline__ void async_b128(uint32_t lds_addr, const char* gaddr) {
    asm volatile("global_load_async_to_lds_b128 %0, %1, off offset:%2 th:TH_LOAD_NT"
                 :: "v"(lds_addr), "v"(gaddr), "n"(OFF)
                 : "memory");
}

__device__ __forceinline__ void async_tile_load(const float* __restrict__ gsrc_row0,
                                                float (*lds_rows)[ROW_ELEMS],
                                                int t) {
    // Each thread copies 16 bytes (4 floats) per row: elems 4t..4t+3.
    const char* g = (const char*)gsrc_row0 + (size_t)t * 16;
    // Low 32 bits of a flat pointer into LDS == wave-relative LDS byte address.
    uint32_t l = (uint32_t)(uintptr_t)(&lds_rows[0][0]) + (uint32_t)t * 16u;
    async_b128<0 * ROW_BYTES>(l, g);
    async_b128<1 * ROW_BYTES>(l, g);
    async_b128<2 * ROW_BYTES>(l, g);
    async_b128<3 * ROW_BYTES>(l, g);
}

__global__ __launch_bounds__(THREADS) void perm_gather_kernel(
    const float* __restrict__ features,
    const int*   __restrict__ perm,
    float*       __restrict__ out) {
    __shared__ __align__(16) float buf[2][ROWS_PER_TILE][ROW_ELEMS];  // 32 KB

    const int t = threadIdx.x;
    const size_t row0 = (size_t)blockIdx.x * ROWS_PER_BLOCK;

    // Per-thread permutation indices for output channels 4t..4t+3 (coalesced
    // b128 load), reused across all 32 rows this block handles.
    const int4 p = ((const int4*)perm)[t];

    const float* gbase = features + row0 * ROW_ELEMS;
    float*       obase = out      + row0 * ROW_ELEMS;

    // Prime the pipeline: tile 0 -> buffer 0.
    async_tile_load(gbase, buf[0], t);

#pragma unroll
    for (int i = 0; i < TILES; ++i) {
        // Prefetch next tile into the other buffer while we drain this one.
        if (i + 1 < TILES) {
            async_tile_load(gbase + (size_t)(i + 1) * ROWS_PER_TILE * ROW_ELEMS,
                            buf[(i + 1) & 1], t);
            // 4 copies (next tile) still in flight; oldest 4 (current tile) done.
            asm volatile("s_wait_asynccnt %0" :: "n"(ROWS_PER_TILE) : "memory");
        } else {
            asm volatile("s_wait_asynccnt 0" ::: "memory");
        }
        __syncthreads();  // all waves' DMA for current tile visible in LDS

        const float* rl0  = &buf[i & 1][0][0];
        float*       orow = obase + (size_t)i * ROWS_PER_TILE * ROW_ELEMS;
#pragma unroll
        for (int r = 0; r < ROWS_PER_TILE; ++r) {
            const float* rl = rl0 + r * ROW_ELEMS;  // LDS row base
            v4f v;
            v.x = rl[p.x];  // ds_load_b32 gathers: 64 LDS banks absorb the scatter
            v.y = rl[p.y];
            v.z = rl[p.z];
            v.w = rl[p.w];
            // Streaming store: global_store_b128 th:TH_STORE_NT (write-once data,
            // keep the 512MB output out of L2).
            __builtin_nontemporal_store(v, (v4f*)(orow + (size_t)r * ROW_ELEMS) + t);
        }
        __syncthreads();  // buffer i&1 fully consumed before its reuse at tile i+2
    }
}

extern "C" void kernel_launch(void* const* d_in, const int* in_sizes, int n_in,
                              void* d_out, int out_size, void* d_ws, size_t ws_size,
                              hipStream_t stream) {
    const float* features = (const float*)d_in[0];
    const int*   perm     = (const int*)d_in[1];
    float*       out      = (float*)d_out;

    const int rows   = in_sizes[0] / ROW_ELEMS;       // 131072
    const int blocks = rows / ROWS_PER_BLOCK;         // 4096 (131072 % 32 == 0)

    perm_gather_kernel<<<dim3(blocks), dim3(THREADS), 0, stream>>>(features, perm, out);
}